// RigidTransformation3DImputation_22007412425391
// MI455X (gfx1250) — compile-verified
//
#include <hip/hip_runtime.h>
#include <math.h>

// ---------------------------------------------------------------------------
// RigidTransformation3DImputation for MI455X (gfx1250, wave32, WMMA)
//
// 3D FFT == 3 passes of (96x96 DFT matrix) x (96 x 9216 data) per batch,
// executed as V_WMMA_F32_16X16X4_F32 tile matmuls (f32 precision end-to-end).
// Twiddle tiles are staged in LDS (shared by all 8 waves of a block); data
// fragments are software-pipelined in registers (prefetch distance 2) so
// WMMAs overlap outstanding global loads instead of draining them.
// ---------------------------------------------------------------------------

typedef __attribute__((ext_vector_type(2))) float v2f;
typedef __attribute__((ext_vector_type(4))) float v4f;
typedef __attribute__((ext_vector_type(8))) float v8f;

#define NN   96
#define NN2  (NN * NN)
#define NN3  (NN * NN * NN)
#define NB   16
#define NTOT (NB * NN3)
#define KSTEPS (NN / 4)   // 24 K-steps of 4

static constexpr float PI_F     = 3.14159265358979323846f;
static constexpr float TWO_PI_F = 6.28318530717958647692f;

// D = A(16x4 f32) * B(4x16 f32) + C(16x16 f32), full-wave op.
__device__ __forceinline__ v8f wmma4(v2f a, v2f b, v8f c) {
  return __builtin_amdgcn_wmma_f32_16x16x4_f32(
      /*neg_a=*/false, a, /*neg_b=*/false, b,
      /*c_mod=*/(short)0, c, /*reuse_a=*/false, /*reuse_b=*/false);
}

// ---------------------------------------------------------------------------
// theta -> 3x4 pose matrices (ZYZ euler, reference convention)
// ---------------------------------------------------------------------------
__global__ void theta_kernel(const float* __restrict__ theta,
                             float* __restrict__ tm) {
  int b = threadIdx.x;
  if (b >= NB) return;
  const float phi = theta[b * 6 + 0] * TWO_PI_F - PI_F;
  const float th  = theta[b * 6 + 1] * TWO_PI_F - PI_F;
  const float psi = theta[b * 6 + 2] * TWO_PI_F - PI_F;
  const float cp = cosf(phi), sp = sinf(phi);
  const float ct = cosf(th),  st = sinf(th);
  const float cs = cosf(psi), ss = sinf(psi);
  float* T = tm + b * 12;
  T[0] =  cs * ct * cp - ss * sp;  T[1] = -cs * ct * sp - ss * cp;  T[2]  = cs * st;  T[3]  = theta[b * 6 + 3];
  T[4] =  ss * ct * cp + cs * sp;  T[5] = -ss * ct * sp + cs * cp;  T[6]  = ss * st;  T[7]  = theta[b * 6 + 4];
  T[8] = -st * cp;                 T[9] =  st * sp;                 T[10] = ct;       T[11] = theta[b * 6 + 5];
}

// ---------------------------------------------------------------------------
// DFT twiddle tables. Inverse tables pre-scaled by 1/96 (3 passes => 1/96^3).
// ---------------------------------------------------------------------------
__global__ void twiddle_kernel(float* __restrict__ Wfr, float* __restrict__ Wfi,
                               float* __restrict__ Wbr, float* __restrict__ Wbi) {
  int idx = blockIdx.x * blockDim.x + threadIdx.x;
  if (idx >= NN2) return;
  int k = idx / NN, n = idx % NN;
  float ang = -(TWO_PI_F / (float)NN) * (float)((k * n) % NN);
  float c = cosf(ang), s = sinf(ang);
  Wfr[idx] = c;
  Wfi[idx] = s;
  Wbr[idx] = c * (1.0f / (float)NN);
  Wbi[idx] = -s * (1.0f / (float)NN);
}

// ---------------------------------------------------------------------------
// Trilinear sample of zero-padded X (98^3 grid, border clamp), interior crop.
// ---------------------------------------------------------------------------
__device__ __forceinline__ float readPad(const float* __restrict__ X,
                                         int b, int z, int y, int x) {
  if ((unsigned)(z - 1) >= (unsigned)NN || (unsigned)(y - 1) >= (unsigned)NN ||
      (unsigned)(x - 1) >= (unsigned)NN)
    return 0.0f;
  return X[(long)b * NN3 + (long)(z - 1) * NN2 + (y - 1) * NN + (x - 1)];
}

__global__ __launch_bounds__(256) void sample_x_kernel(
    const float* __restrict__ X, const float* __restrict__ tm,
    float* __restrict__ Xt) {
  int idx = blockIdx.x * blockDim.x + threadIdx.x;
  if (idx >= NTOT) return;
  int b = idx / NN3;
  int r = idx - b * NN3;
  int d = r / NN2;
  int h = (r / NN) % NN;
  int w = r % NN;
  const float* T = tm + b * 12;
  // interior point (d+1,h+1,w+1) of linspace(-1,1,98)
  const float xn = -1.0f + (2.0f / 97.0f) * (float)(w + 1);
  const float yn = -1.0f + (2.0f / 97.0f) * (float)(h + 1);
  const float zn = -1.0f + (2.0f / 97.0f) * (float)(d + 1);
  const float gx = T[0] * xn + T[1] * yn + T[2]  * zn + T[3];
  const float gy = T[4] * xn + T[5] * yn + T[6]  * zn + T[7];
  const float gz = T[8] * xn + T[9] * yn + T[10] * zn + T[11];
  // unnormalize onto padded 98 grid, clamp to border [0,97]
  float ix = fminf(fmaxf((gx + 1.0f) * 48.5f, 0.0f), 97.0f);
  float iy = fminf(fmaxf((gy + 1.0f) * 48.5f, 0.0f), 97.0f);
  float iz = fminf(fmaxf((gz + 1.0f) * 48.5f, 0.0f), 97.0f);
  float x0f = floorf(ix), y0f = floorf(iy), z0f = floorf(iz);
  float wx = ix - x0f, wy = iy - y0f, wz = iz - z0f;
  int x0 = (int)x0f, y0 = (int)y0f, z0 = (int)z0f;
  int x1 = min(x0 + 1, 97), y1 = min(y0 + 1, 97), z1 = min(z0 + 1, 97);
  float c000 = readPad(X, b, z0, y0, x0), c001 = readPad(X, b, z0, y0, x1);
  float c010 = readPad(X, b, z0, y1, x0), c011 = readPad(X, b, z0, y1, x1);
  float c100 = readPad(X, b, z1, y0, x0), c101 = readPad(X, b, z1, y0, x1);
  float c110 = readPad(X, b, z1, y1, x0), c111 = readPad(X, b, z1, y1, x1);
  float l00 = c000 + wx * (c001 - c000);
  float l01 = c010 + wx * (c011 - c010);
  float l10 = c100 + wx * (c101 - c100);
  float l11 = c110 + wx * (c111 - c110);
  float l0 = l00 + wy * (l01 - l00);
  float l1 = l10 + wy * (l11 - l10);
  Xt[idx] = l0 + wz * (l1 - l0);
}

// ---------------------------------------------------------------------------
// One 1D DFT pass along AXIS as WMMA tile-matmul: Out[k,col] = W[k,:]·V[:,col]
// Each wave owns a 16(freq) x 16(col) complex output tile; 8 waves per block
// share one 16x96 complex twiddle tile staged in LDS.
//   HAS_II: input has imaginary part;  HAS_OI: output needs imaginary part.
// AXIS 0: stride 1    cols=(z,y)   AXIS 1: stride 96  cols=(z,x)
// AXIS 2: stride 96^2 cols=(y,x)
// ---------------------------------------------------------------------------
template <int AXIS, bool HAS_II, bool HAS_OI>
__global__ __launch_bounds__(256) void dft_pass(
    const float* __restrict__ inR, const float* __restrict__ inI,
    const float* __restrict__ Wr, const float* __restrict__ Wi,
    float* __restrict__ outR, float* __restrict__ outI) {
  constexpr int S = (AXIS == 0) ? 1 : (AXIS == 1) ? NN : NN2;

  __shared__ float lWr[16 * NN];
  __shared__ float lWi[16 * NN];

  const int lane = threadIdx.x & 31;
  const int wave = threadIdx.x >> 5;
  const int ct = blockIdx.x * 8 + wave;     // 576 col tiles / 8 waves
  const int kt = blockIdx.y;                // 6 freq tiles
  const int b  = blockIdx.z;                // 16 batches
  const int c  = ct * 16 + (lane & 15);     // this lane's column (N dim)

  // Stage this block's 16x96 twiddle rows (shared by all 8 waves).
  for (int i = threadIdx.x; i < 16 * NN; i += 256) {
    lWr[i] = Wr[kt * 16 * NN + i];
    lWi[i] = Wi[kt * 16 * NN + i];
  }
  __syncthreads();

  int cm;
  if constexpr (AXIS == 0)      cm = c * NN;
  else if constexpr (AXIS == 1) cm = (c / NN) * NN2 + (c % NN);
  else                          cm = c;

  const long base = (long)b * NN3 + cm;

  // A fragment (16x4 f32): lane L -> row M = L&15, K = (L>>4)*2 + reg.
  // B fragment (4x16 f32): reg r, lane-half h -> K = 2h + r, N = L&15.
  const int m  = lane & 15;
  const int kk = (lane >> 4) * 2;
  const float* wr = lWr + m * NN + kk;   // LDS, 8B-aligned
  const float* wi = lWi + m * NN + kk;
  const float* pR = inR + base + (long)kk * S;
  const float* pI = HAS_II ? (inI + base + (long)kk * S) : nullptr;

  auto loadB = [&](const float* p, int n) -> v2f {
    v2f r;
    if constexpr (AXIS == 0) {
      r = *(const v2f*)(p + n);            // contiguous -> global_load_b64
    } else {
      r.x = p[(long)n * S];
      r.y = p[(long)(n + 1) * S];
    }
    return r;
  };

  v8f arr = {};  // sum Wr*Vr
  v8f aii = {};  // sum Wi*Vi
  v8f ari = {};  // sum Wr*Vi
  v8f air = {};  // sum Wi*Vr

  // Software pipeline: B prefetch distance 2 (global), A distance 1 (LDS).
  constexpr int PF = 2;
  v2f BrBuf[PF], BiBuf[PF];
#pragma unroll
  for (int j = 0; j < PF; ++j) {
    BrBuf[j] = loadB(pR, 4 * j);
    if (HAS_II) BiBuf[j] = loadB(pI, 4 * j);
  }
  v2f Ar_c = *(const v2f*)(wr);
  v2f Ai_c = *(const v2f*)(wi);

#pragma unroll
  for (int t = 0; t < KSTEPS; ++t) {
    const v2f Br_c = BrBuf[t % PF];
    v2f Bi_c;
    if (HAS_II) Bi_c = BiBuf[t % PF];
    if (t + PF < KSTEPS) {
      BrBuf[t % PF] = loadB(pR, 4 * (t + PF));
      if (HAS_II) BiBuf[t % PF] = loadB(pI, 4 * (t + PF));
    }
    v2f Ar_n = Ar_c, Ai_n = Ai_c;
    if (t + 1 < KSTEPS) {
      Ar_n = *(const v2f*)(wr + 4 * (t + 1));
      Ai_n = *(const v2f*)(wi + 4 * (t + 1));
    }
    arr = wmma4(Ar_c, Br_c, arr);
    if (HAS_OI) air = wmma4(Ai_c, Br_c, air);
    if (HAS_II) {
      aii = wmma4(Ai_c, Bi_c, aii);
      if (HAS_OI) ari = wmma4(Ar_c, Bi_c, ari);
    }
    Ar_c = Ar_n;
    Ai_c = Ai_n;
  }

  // C/D layout: VGPR v, lanes 0-15 -> M=v, lanes 16-31 -> M=v+8, N=lane&15.
  const int rbase = kt * 16 + (lane >> 4) * 8;
  float* oR = outR + base + (long)rbase * S;
  float* oI = HAS_OI ? (outI + base + (long)rbase * S) : nullptr;

  float re[8], im[8];
#pragma unroll
  for (int v = 0; v < 8; ++v) {
    re[v] = HAS_II ? (arr[v] - aii[v]) : arr[v];
    if (HAS_OI) im[v] = HAS_II ? (air[v] + ari[v]) : air[v];
  }

  if constexpr (AXIS == 0) {
    // 8 contiguous floats per lane, 32B-aligned -> two global_store_b128
    v4f r0 = {re[0], re[1], re[2], re[3]};
    v4f r1 = {re[4], re[5], re[6], re[7]};
    *(v4f*)(oR)     = r0;
    *(v4f*)(oR + 4) = r1;
    if (HAS_OI) {
      v4f i0 = {im[0], im[1], im[2], im[3]};
      v4f i1 = {im[4], im[5], im[6], im[7]};
      *(v4f*)(oI)     = i0;
      *(v4f*)(oI + 4) = i1;
    }
  } else {
#pragma unroll
    for (int v = 0; v < 8; ++v) {
      oR[(long)v * S] = re[v];
      if (HAS_OI) oI[(long)v * S] = im[v];
    }
  }
}

// ---------------------------------------------------------------------------
// Spectral combine with fused nearest-neighbor (zeros) mask resampling:
//   FT = FX * M1_t + FY * M2_t   (masks are real)
// ---------------------------------------------------------------------------
__global__ __launch_bounds__(256) void combine_kernel(
    const float* __restrict__ fxr, const float* __restrict__ fxi,
    const float* __restrict__ fyr, const float* __restrict__ fyi,
    const float* __restrict__ m1, const float* __restrict__ m2,
    const float* __restrict__ tm,
    float* __restrict__ ftr, float* __restrict__ fti) {
  int idx = blockIdx.x * blockDim.x + threadIdx.x;
  if (idx >= NTOT) return;
  int b = idx / NN3;
  int r = idx - b * NN3;
  int d = r / NN2;
  int h = (r / NN) % NN;
  int w = r % NN;
  const float* T = tm + b * 12;
  const float xn = -1.0f + (2.0f / 95.0f) * (float)w;
  const float yn = -1.0f + (2.0f / 95.0f) * (float)h;
  const float zn = -1.0f + (2.0f / 95.0f) * (float)d;
  const float gx = T[0] * xn + T[1] * yn + T[2]  * zn + T[3];
  const float gy = T[4] * xn + T[5] * yn + T[6]  * zn + T[7];
  const float gz = T[8] * xn + T[9] * yn + T[10] * zn + T[11];
  // nearest (ties-to-even like jnp.round), zeros outside
  const float fx = rintf((gx + 1.0f) * 47.5f);
  const float fy = rintf((gy + 1.0f) * 47.5f);
  const float fz = rintf((gz + 1.0f) * 47.5f);
  const bool valid = (fx >= 0.0f) && (fx <= 95.0f) && (fy >= 0.0f) &&
                     (fy <= 95.0f) && (fz >= 0.0f) && (fz <= 95.0f);
  float M1 = 0.0f, M2 = 0.0f;
  if (valid) {
    long a = (long)b * NN3 + (long)(int)fz * NN2 + (int)fy * NN + (int)fx;
    M1 = m1[a];
    M2 = m2[a];
  }
  ftr[idx] = fxr[idx] * M1 + fyr[idx] * M2;
  fti[idx] = fxi[idx] * M1 + fyi[idx] * M2;
}

// ---------------------------------------------------------------------------
// Launcher. Workspace layout (floats):
//   [0,256)      theta matrices (16 x 12)
//   [256,37120)  twiddle tables Wfr,Wfi,Wbr,Wbi (4 x 9216)
//   then 6 buffers of 16*96^3 = P.r P.i Q.r Q.i R.r R.i  (~340 MB total)
// d_out doubles as X_t scratch (fully consumed before the final pass).
// ---------------------------------------------------------------------------
extern "C" void kernel_launch(void* const* d_in, const int* in_sizes, int n_in,
                              void* d_out, int out_size, void* d_ws,
                              size_t ws_size, hipStream_t stream) {
  const float* X     = (const float*)d_in[0];
  const float* Y     = (const float*)d_in[1];
  const float* m1    = (const float*)d_in[2];
  const float* m2    = (const float*)d_in[3];
  const float* theta = (const float*)d_in[4];
  float* out = (float*)d_out;

  float* ws  = (float*)d_ws;
  float* tm  = ws;
  float* Wfr = ws + 256;
  float* Wfi = Wfr + NN2;
  float* Wbr = Wfi + NN2;
  float* Wbi = Wbr + NN2;
  size_t off = 256 + 4 * (size_t)NN2;
  const size_t NT = (size_t)NTOT;
  float* Pr = ws + off;        off += NT;
  float* Pi = ws + off;        off += NT;
  float* Qr = ws + off;        off += NT;
  float* Qi = ws + off;        off += NT;
  float* Rr = ws + off;        off += NT;
  float* Ri = ws + off;        off += NT;

  theta_kernel<<<1, 32, 0, stream>>>(theta, tm);
  twiddle_kernel<<<(NN2 + 255) / 256, 256, 0, stream>>>(Wfr, Wfi, Wbr, Wbi);

  // X_t = trilinear-border resample of padded X  (into d_out as scratch)
  sample_x_kernel<<<NTOT / 256, 256, 0, stream>>>(X, tm, out);

  const dim3 dg(72, 6, 16);  // 576/8 col-tile groups, 6 freq tiles, 16 batches
  const dim3 db(256);        // 8 waves per block

  // FX = fftn(X_t):  out -> P -> Q -> P
  dft_pass<0, false, true><<<dg, db, 0, stream>>>(out, nullptr, Wfr, Wfi, Pr, Pi);
  dft_pass<1, true,  true><<<dg, db, 0, stream>>>(Pr, Pi, Wfr, Wfi, Qr, Qi);
  dft_pass<2, true,  true><<<dg, db, 0, stream>>>(Qr, Qi, Wfr, Wfi, Pr, Pi);
  // FY = fftn(Y):  Y -> Q -> R -> Q
  dft_pass<0, false, true><<<dg, db, 0, stream>>>(Y, nullptr, Wfr, Wfi, Qr, Qi);
  dft_pass<1, true,  true><<<dg, db, 0, stream>>>(Qr, Qi, Wfr, Wfi, Rr, Ri);
  dft_pass<2, true,  true><<<dg, db, 0, stream>>>(Rr, Ri, Wfr, Wfi, Qr, Qi);
  // FT = FX*M1_t + FY*M2_t  (masks resampled on the fly) -> R
  combine_kernel<<<NTOT / 256, 256, 0, stream>>>(Pr, Pi, Qr, Qi, m1, m2, tm,
                                                 Rr, Ri);
  // out = ifftn(FT).real : R -> P -> R -> out  (Wb pre-scaled by 1/96/pass)
  dft_pass<0, true, true ><<<dg, db, 0, stream>>>(Rr, Ri, Wbr, Wbi, Pr, Pi);
  dft_pass<1, true, true ><<<dg, db, 0, stream>>>(Pr, Pi, Wbr, Wbi, Rr, Ri);
  dft_pass<2, true, false><<<dg, db, 0, stream>>>(Rr, Ri, Wbr, Wbi, out, nullptr);
}